// DGCFModel_4887672782966
// MI455X (gfx1250) — compile-verified
//
#include <hip/hip_runtime.h>

#define N_USERS  100000
#define N_ITEMS  50000
#define N_NODES  150000   // N_USERS + N_ITEMS
#define N_EDGES  1000000
#define EMBED_K  64       // 4 intents x 16 chunk, row-major [N, 64]

typedef __attribute__((ext_vector_type(2))) float v2f;
typedef __attribute__((ext_vector_type(8))) float v8f;

// Non-returning relaxed agent-scope fp32 atomic add -> global_atomic_add_f32
__device__ __forceinline__ void atomAddF32(float* p, float v) {
    (void)__hip_atomic_fetch_add(p, v, __ATOMIC_RELAXED, __HIP_MEMORY_SCOPE_AGENT);
}

__global__ void k_zero(float* __restrict__ p, int n) {
    int i = blockIdx.x * blockDim.x + threadIdx.x;
    if (i < n) p[i] = 0.0f;
}

__global__ void k_init_x(const float* __restrict__ Gu, const float* __restrict__ Gi,
                         float* __restrict__ x) {
    int i = blockIdx.x * blockDim.x + threadIdx.x;
    const int total  = N_NODES * EMBED_K;
    const int uElems = N_USERS * EMBED_K;
    if (i >= total) return;
    x[i] = (i < uElems) ? Gu[i] : Gi[i - uElems];
}

// Per-edge softmax over 4 intents; accumulate degree at both endpoints.
__global__ void k_deg(const int* __restrict__ row, const int* __restrict__ col,
                      const float* __restrict__ intents, float* __restrict__ deg) {
    int e = blockIdx.x * blockDim.x + threadIdx.x;
    if (e >= N_EDGES) return;
    float z0 = intents[e];
    float z1 = intents[N_EDGES + e];
    float z2 = intents[2 * N_EDGES + e];
    float z3 = intents[3 * N_EDGES + e];
    float m  = fmaxf(fmaxf(z0, z1), fmaxf(z2, z3));
    float w0 = __expf(z0 - m), w1 = __expf(z1 - m);
    float w2 = __expf(z2 - m), w3 = __expf(z3 - m);
    float inv = 1.0f / (w0 + w1 + w2 + w3);
    w0 *= inv; w1 *= inv; w2 *= inv; w3 *= inv;
    int r = row[e], c = col[e];
    atomAddF32(deg + r * 4 + 0, w0); atomAddF32(deg + r * 4 + 1, w1);
    atomAddF32(deg + r * 4 + 2, w2); atomAddF32(deg + r * 4 + 3, w3);
    atomAddF32(deg + c * 4 + 0, w0); atomAddF32(deg + c * 4 + 1, w1);
    atomAddF32(deg + c * 4 + 2, w2); atomAddF32(deg + c * 4 + 3, w3);
}

// In-place deg -> deg^{-1/2}, with deg<=0 -> 0 (matches reference's inf->0).
__global__ void k_dis(float* __restrict__ deg) {
    int i = blockIdx.x * blockDim.x + threadIdx.x;
    if (i >= N_NODES * 4) return;
    float d = deg[i];
    deg[i] = (d > 0.0f) ? rsqrtf(fmaxf(d, 1e-12f)) : 0.0f;
}

// One wave32 per 16-edge block. For each intent k (16-elem chunk):
//   D[16 edges x 16 elems] = diag(norm[.,k]) * Xgathered   via 4x V_WMMA_F32_16X16X4_F32
// then scatter-add rows of D to x_dst[col[e]] with fp32 atomics.
__global__ __launch_bounds__(128) void k_propagate(
        const float* __restrict__ x_src, float* __restrict__ x_dst,
        const float* __restrict__ dis,
        const int* __restrict__ row, const int* __restrict__ col) {
    const int wave = blockIdx.x * (blockDim.x >> 5) + (threadIdx.x >> 5);
    const int lane = threadIdx.x & 31;
    const int hl   = lane & 15;            // position within half-wave
    const bool hi  = lane >= 16;
    const int base = wave * 16;            // 16 edges per wave; N_EDGES % 16 == 0

    // A-operand edge for this lane (A layout: M = lane&15 in both half-waves)
    const int eA  = base + hl;
    const int irA = row[eA];
    const int icA = col[eA];
    __builtin_prefetch(&x_src[irA * 64], 0, 0);   // global_prefetch_b8 of source row
    const float4 dr = *(const float4*)(dis + irA * 4);
    const float4 dc = *(const float4*)(dis + icA * 4);
    float nrm[4];
    nrm[0] = dr.x * dc.x; nrm[1] = dr.y * dc.y;
    nrm[2] = dr.z * dc.z; nrm[3] = dr.w * dc.w;

    // D layout: component v holds M = v (+8 in hi half); cache those cols
    int colv[8];
#pragma unroll
    for (int v = 0; v < 8; ++v) colv[v] = col[base + v + (hi ? 8 : 0)];

    const int kx = hi ? 2 : 0;             // K indices {kx, kx+1} live in this half-wave

#pragma unroll
    for (int k = 0; k < 4; ++k) {          // intent
        v8f acc = {};
#pragma unroll
        for (int t = 0; t < 4; ++t) {      // rank-4 slice of the 16x16 diagonal
            v2f a, b;
            // A_t[M][kk] = (M == 4t+kk) ? norm[e_M, k] : 0
            a.x = (hl == 4 * t + kx)     ? nrm[k] : 0.0f;
            a.y = (hl == 4 * t + kx + 1) ? nrm[k] : 0.0f;
            // B_t[kk][N] = x_src[row[e_{4t+kk}]][16k + N]
            const int rb0 = row[base + 4 * t + kx];
            const int rb1 = row[base + 4 * t + kx + 1];
            b.x = x_src[rb0 * 64 + k * 16 + hl];
            b.y = x_src[rb1 * 64 + k * 16 + hl];
            acc = __builtin_amdgcn_wmma_f32_16x16x4_f32(
                      false, a, false, b, (short)0, acc, false, false);
        }
        // scatter: D[M][N] -> x_dst[col[e_M]*64 + 16k + N]
#pragma unroll
        for (int v = 0; v < 8; ++v)
            atomAddF32(x_dst + colv[v] * 64 + k * 16 + hl, acc[v]);
    }
}

extern "C" void kernel_launch(void* const* d_in, const int* in_sizes, int n_in,
                              void* d_out, int out_size, void* d_ws, size_t ws_size,
                              hipStream_t stream) {
    const float* Gu      = (const float*)d_in[0];
    const float* Gi      = (const float*)d_in[1];
    const int*   edge    = (const int*)d_in[2];      // [2, E]
    const float* intents = (const float*)d_in[3];    // [4, E]
    const int* row = edge;
    const int* col = edge + N_EDGES;
    float* out = (float*)d_out;                      // [N, 64]

    float* deg  = (float*)d_ws;                      // [N, 4]  (becomes dis in place)
    float* xbuf = deg + (size_t)N_NODES * 4;         // [N, 64] ping-pong partner of d_out

    const int NX = N_NODES * EMBED_K;                // 9,600,000
    const int ND = N_NODES * 4;                      // 600,000
    dim3 b256(256);
    const int waves  = N_EDGES / 16;                 // 62,500
    dim3 pgrid(waves / 4), pblock(128);              // 4 waves/block, exact cover

    // degree -> dis (layer-invariant)
    k_zero<<<dim3((ND + 255) / 256), b256, 0, stream>>>(deg, ND);
    k_deg<<<dim3((N_EDGES + 255) / 256), b256, 0, stream>>>(row, col, intents, deg);
    k_dis<<<dim3((ND + 255) / 256), b256, 0, stream>>>(deg);

    // x0 in ws; layer outputs alternate out -> ws -> out
    k_init_x<<<dim3((NX + 255) / 256), b256, 0, stream>>>(Gu, Gi, xbuf);

    k_zero<<<dim3((NX + 255) / 256), b256, 0, stream>>>(out, NX);
    k_propagate<<<pgrid, pblock, 0, stream>>>(xbuf, out, deg, row, col);   // layer 1

    k_zero<<<dim3((NX + 255) / 256), b256, 0, stream>>>(xbuf, NX);
    k_propagate<<<pgrid, pblock, 0, stream>>>(out, xbuf, deg, row, col);   // layer 2

    k_zero<<<dim3((NX + 255) / 256), b256, 0, stream>>>(out, NX);
    k_propagate<<<pgrid, pblock, 0, stream>>>(xbuf, out, deg, row, col);   // layer 3
}